// TitanMAC_34668976013741
// MI455X (gfx1250) — compile-verified
//
#include <hip/hip_runtime.h>
#include <hip/hip_bf16.h>

typedef __attribute__((ext_vector_type(2))) float v2f;
typedef __attribute__((ext_vector_type(8))) float v8f;

namespace {

constexpr long Bt   = 2;
constexpr long Lq   = 2048;
constexpr long Dm   = 512;
constexpr long Hh   = 8;
constexpr long CHc  = 256;
constexpr long NPp  = 16;
constexpr long MHh  = 1024;
constexpr long Vv   = 32000;
constexpr long Tt   = NPp + 2 * CHc;   // 528
constexpr long NC   = Lq / CHc;        // 8 chunks
constexpr long SEQ  = NC * Tt;         // 4224
constexpr long TAIL = SEQ - Lq;        // 2176 (skip rows before last L positions)
constexpr long PSTR = Dm * MHh + MHh + MHh * Dm + Dm;  // per-batch param block
constexpr long PO_W1 = 0;
constexpr long PO_B1 = Dm * MHh;
constexpr long PO_W2 = PO_B1 + MHh;
constexpr long PO_B2 = PO_W2 + MHh * Dm;

// ---------------- workspace layout (floats) ----------------
constexpr long OFF_X   = 0;
constexpr long OFF_PAR = OFF_X   + Bt * Lq * Dm;
constexpr long OFF_SUR = OFF_PAR + Bt * PSTR;
constexpr long OFF_GRD = OFF_SUR + Bt * PSTR;
constexpr long OFF_QM  = OFF_GRD + Bt * PSTR;
constexpr long OFF_HR  = OFF_QM  + Bt * CHc * Dm;
constexpr long OFF_HT  = OFF_HR  + Bt * CHc * MHh;
constexpr long OFF_TT  = OFF_HT  + Bt * CHc * Dm;
constexpr long OFF_QKV = OFF_TT  + Bt * Tt * Dm;
constexpr long OFF_AO  = OFF_QKV + Bt * Tt * 3 * Dm;
constexpr long OFF_KM  = OFF_AO  + Bt * Tt * Dm;
constexpr long OFF_VM  = OFF_KM  + Bt * Tt * Dm;
constexpr long OFF_H   = OFF_VM  + Bt * Tt * Dm;
constexpr long OFF_A   = OFF_H   + Bt * Tt * MHh;
constexpr long OFF_R   = OFF_A   + Bt * Tt * MHh;
constexpr long OFF_GA  = OFF_R   + Bt * Tt * Dm;
constexpr long OFF_QY  = OFF_GA  + Bt * Tt * MHh;
constexpr long OFF_HY  = OFF_QY  + Bt * Tt * Dm;
constexpr long OFF_YT  = OFF_HY  + Bt * Tt * MHh;
constexpr long OFF_MT  = OFF_YT  + Bt * SEQ * Dm;
constexpr long OFF_GZ  = OFF_MT  + Bt * SEQ * Dm;

__device__ __forceinline__ void wait_asynccnt0() {
#if __has_builtin(__builtin_amdgcn_s_wait_asynccnt)
  __builtin_amdgcn_s_wait_asynccnt(0);
#else
  asm volatile("s_wait_asynccnt 0x0" ::: "memory");
#endif
}

__device__ __forceinline__ void async_copy_b64(unsigned ldsAddr, const float* gaddr) {
  asm volatile("global_load_async_to_lds_b64 %0, %1, off"
               :: "v"(ldsAddr), "v"(gaddr) : "memory");
}

// ============================================================
// fp32 WMMA GEMM: C = act(op(A) @ op(B) + bias)
//   V_WMMA_F32_16X16X4_F32, wave32.
// Block = 128 threads = 4 waves; block tile = 16(M) x 256(N);
// each wave owns a 16x64 strip (4 accumulators -> 4 WMMA per A frag).
// A tile (16 x 16 f32 = 1KB) is staged into a DOUBLE-BUFFERED LDS slab
// with GLOBAL_LOAD_ASYNC_TO_LDS_B64 (ASYNCcnt): stage i+1's copy is
// issued before stage i's math, hidden behind the 16 B-fragment global
// loads + 16 WMMAs of the stage. The end-of-stage barrier guarantees no
// wave still reads the buffer that the next iteration overwrites.
// All 16 B frags of a stage are hoisted into VGPRs first so only one
// loadcnt wait gates the WMMA chain; B rows are walked with pointer
// increments (no per-step 64-bit multiplies).
// Shapes launched: M%16==0, N%256==0, K%16==0 (528 = 33*16). EXEC stays
// all-ones (guards are wave-uniform) as WMMA requires.
// Fragment layouts per ISA 7.12.2:
//   A frag (16x4): lane<16 -> row m0+lid holds K=k0,k0+1; lanes>=16 -> K=k0+2,k0+3
//   B frag (4x16): lane<16 -> col n0+lid rows K=k0,k0+1; lanes>=16 -> K=k0+2,k0+3
//   C/D: vgpr r, lane<16 -> (m0+r, n0+lid); lane>=16 -> (m0+r+8, n0+lid)
// ============================================================
template <int TA, int TB, int ACT>
__global__ __launch_bounds__(128) void gemm_wmma_f32(
    const float* __restrict__ A, int lda, long strideA,
    const float* __restrict__ B, int ldb, long strideB,
    const float* __restrict__ bias, long strideBias,
    float* __restrict__ C, int ldc, long strideC,
    int M, int N, int K)
{
  __shared__ float sA[2][16 * 16];   // double-buffered; TA=0: [m][k]; TA=1: [k][m]

  const int bz = blockIdx.z;
  A += (size_t)bz * strideA;
  B += (size_t)bz * strideB;
  C += (size_t)bz * strideC;
  if (bias) bias += (size_t)bz * strideBias;

  const int tid  = threadIdx.x;
  const int wave = tid >> 5;
  const int lane = tid & 31;
  const int half = lane >> 4;
  const int lid  = lane & 15;

  const int m0 = blockIdx.y * 16;
  const int n0 = blockIdx.x * 256 + wave * 64;
  const int cb = n0 + lid;               // base B/C column for this lane

  // ---- async staging plan: 128 threads x 8B = 1KB = one 16x16 f32 tile ----
  const int srow = tid >> 3;             // 0..15 (m-row for TA=0, k-row for TA=1)
  const int scol = (tid & 7) * 2;        // 0..14 (even float index)
  const unsigned ldsOff = (unsigned)(srow * 16 + scol) * 4u;
  const unsigned ldsB0 = (unsigned)(uintptr_t)(void*)&sA[0][0] + ldsOff;
  const unsigned ldsB1 = (unsigned)(uintptr_t)(void*)&sA[1][0] + ldsOff;
  const float* gstage = TA ? (A + (size_t)srow * lda + m0 + scol)
                           : (A + (size_t)(m0 + srow) * lda + scol);
  const size_t gstep = TA ? (size_t)16 * lda : (size_t)16;

  // ---- B stream pointers (advanced, never re-derived) ----
  const float* b0 = nullptr;   // TB=0: row k0+2*half of B[K x N], at column cb
  const float* b1 = nullptr;   // TB=0: row k0+2*half+1
  const float* bq[4] = {nullptr, nullptr, nullptr, nullptr};  // TB=1 per-subtile rows
  if (TB) {
    #pragma unroll
    for (int s = 0; s < 4; ++s) bq[s] = B + (size_t)(cb + 16 * s) * ldb + 2 * half;
  } else {
    b0 = B + cb + (size_t)(2 * half) * ldb;
    b1 = b0 + ldb;
  }

  v8f acc[4] = {v8f{}, v8f{}, v8f{}, v8f{}};
  const int nstages = K / 16;

  // prologue: fill buffer 0
  async_copy_b64(ldsB0, gstage);
  gstage += gstep;
  wait_asynccnt0();
  __syncthreads();

  for (int st = 0; st < nstages; ++st) {
    // kick off the NEXT A stage into the other LDS buffer (overlapped)
    if (st + 1 < nstages) {
      async_copy_b64((st & 1) ? ldsB0 : ldsB1, gstage);
      gstage += gstep;
    }

    // hoist the stage's 16 B fragments into registers
    v2f bf[4][4];
    if (TB) {
      #pragma unroll
      for (int k4 = 0; k4 < 4; ++k4) {
        #pragma unroll
        for (int s = 0; s < 4; ++s) bf[k4][s] = *(const v2f*)(bq[s] + k4 * 4);
      }
      #pragma unroll
      for (int s = 0; s < 4; ++s) {
        bq[s] += 16;
        __builtin_prefetch(bq[s], 0, 0);   // next stage (L2-resident stream)
      }
    } else {
      const float* r0 = b0;
      const float* r1 = b1;
      #pragma unroll
      for (int k4 = 0; k4 < 4; ++k4) {
        #pragma unroll
        for (int s = 0; s < 4; ++s) { bf[k4][s].x = r0[16 * s]; bf[k4][s].y = r1[16 * s]; }
        r0 += (size_t)4 * ldb;
        r1 += (size_t)4 * ldb;
      }
      b0 += (size_t)16 * ldb;
      b1 += (size_t)16 * ldb;
      __builtin_prefetch(b0, 0, 0);        // next stage (L2-resident stream)
    }

    // 16 WMMAs from the current LDS buffer
    const float* sAb = sA[st & 1];
    #pragma unroll
    for (int k4 = 0; k4 < 4; ++k4) {
      const int ka = 4 * k4 + 2 * half;
      v2f a;
      if (TA) { a.x = sAb[ka * 16 + lid];  a.y = sAb[(ka + 1) * 16 + lid]; }
      else    { a.x = sAb[lid * 16 + ka];  a.y = sAb[lid * 16 + ka + 1]; }
      #pragma unroll
      for (int s = 0; s < 4; ++s) {
        acc[s] = __builtin_amdgcn_wmma_f32_16x16x4_f32(
            false, a, false, bf[k4][s], (short)0, acc[s], false, false);
      }
    }

    // next buffer fully written (own wave) + all waves done with both buffers
    wait_asynccnt0();
    __syncthreads();
  }

  #pragma unroll
  for (int s = 0; s < 4; ++s) {
    const int col = cb + 16 * s;
    const float bv = bias ? bias[col] : 0.0f;
    #pragma unroll
    for (int r = 0; r < 8; ++r) {
      const int row = m0 + r + 8 * half;
      float v = acc[s][r] + bv;
      if (ACT == 1) v = v * (1.0f / (1.0f + __expf(-v)));  // SiLU
      if (ACT == 2) v = 1.0f / (1.0f + __expf(-v));        // sigmoid
      C[(size_t)row * ldc + col] = v;
    }
  }
}

// ---------------- embedding gather: X = tok_emb[tokens] + pos_emb ----------------
__global__ void embed_kernel(const int* __restrict__ tokens,
                             const float* __restrict__ tok_emb,
                             const float* __restrict__ pos_emb,
                             float* __restrict__ X)
{
  const int i = blockIdx.x;   // position
  const int b = blockIdx.y;   // batch
  const long tok = tokens[(size_t)b * Lq + i];
  float* dst = X + (size_t)b * Lq * Dm + (size_t)i * Dm;
  const float* te = tok_emb + (size_t)tok * Dm;
  const float* pe = pos_emb + (size_t)i * Dm;
  for (int d = threadIdx.x; d < Dm; d += blockDim.x) dst[d] = te[d] + pe[d];
}

// ---------------- t = concat([P; ht; chunk_x]) ----------------
__global__ void concat_kernel(float* __restrict__ tt,
                              const float* __restrict__ P,
                              const float* __restrict__ ht,
                              const float* __restrict__ X, int c)
{
  const long idx = (long)blockIdx.x * blockDim.x + threadIdx.x;
  if (idx >= Bt * Tt * Dm) return;
  const long b = idx / (Tt * Dm);
  const long rem = idx - b * Tt * Dm;
  const long row = rem / Dm, col = rem - row * Dm;
  float v;
  if (row < NPp)            v = P[row * Dm + col];
  else if (row < NPp + CHc) v = ht[b * CHc * Dm + (row - NPp) * Dm + col];
  else                      v = X[b * Lq * Dm + (long)c * CHc * Dm + (row - NPp - CHc) * Dm + col];
  tt[idx] = v;
}

// ---------------- multi-head attention over T=528, one (row, head, batch) / block ----------------
__global__ void attn_kernel(const float* __restrict__ qkv, float* __restrict__ out)
{
  __shared__ float qs[64];
  __shared__ float sc[Tt];
  __shared__ float red[64];
  const int row = blockIdx.x;
  const int h   = blockIdx.y;
  const int b   = blockIdx.z;
  const int t   = threadIdx.x;
  const float* base = qkv + (size_t)b * (Tt * 3 * Dm);

  qs[t] = base[(size_t)row * 3 * Dm + h * 64 + t];
  __syncthreads();

  for (int j = t; j < Tt; j += 64) {
    const float* kr = base + (size_t)j * 3 * Dm + Dm + h * 64;
    float s = 0.0f;
    #pragma unroll
    for (int d = 0; d < 64; ++d) s += qs[d] * kr[d];
    sc[j] = s * 0.125f;  // 1/sqrt(64)
  }
  __syncthreads();

  float m = -3.4e38f;
  for (int j = t; j < Tt; j += 64) m = fmaxf(m, sc[j]);
  red[t] = m; __syncthreads();
  for (int s2 = 32; s2 > 0; s2 >>= 1) { if (t < s2) red[t] = fmaxf(red[t], red[t + s2]); __syncthreads(); }
  const float mx = red[0];
  __syncthreads();

  float ls = 0.0f;
  for (int j = t; j < Tt; j += 64) { const float e = __expf(sc[j] - mx); sc[j] = e; ls += e; }
  red[t] = ls; __syncthreads();
  for (int s2 = 32; s2 > 0; s2 >>= 1) { if (t < s2) red[t] += red[t + s2]; __syncthreads(); }
  const float inv = 1.0f / red[0];
  __syncthreads();

  float acc = 0.0f;
  for (int j = 0; j < Tt; ++j)
    acc += sc[j] * base[(size_t)j * 3 * Dm + 2 * Dm + h * 64 + t];
  out[(size_t)b * (Tt * Dm) + (size_t)row * Dm + h * 64 + t] = acc * inv;
}

// ---------------- elementwise kernels ----------------
__global__ void zero_kernel(float* p, long n) {
  const long i = (long)blockIdx.x * blockDim.x + threadIdx.x;
  if (i < n) p[i] = 0.0f;
}
__global__ void silu_kernel(float* __restrict__ a, const float* __restrict__ h, long n) {
  const long i = (long)blockIdx.x * blockDim.x + threadIdx.x;
  if (i < n) { const float x = h[i]; a[i] = x * (1.0f / (1.0f + __expf(-x))); }
}
__global__ void resid2_kernel(float* __restrict__ r, const float* __restrict__ v, long n) {
  const long i = (long)blockIdx.x * blockDim.x + threadIdx.x;
  if (i < n) r[i] = 2.0f * (r[i] - v[i]);
}
__global__ void dsilu_mul_kernel(float* __restrict__ ga, const float* __restrict__ h, long n) {
  const long i = (long)blockIdx.x * blockDim.x + threadIdx.x;
  if (i < n) {
    const float x = h[i];
    const float sg = 1.0f / (1.0f + __expf(-x));
    ga[i] *= sg * (1.0f + x * (1.0f - sg));  // d/dx silu(x)
  }
}
// Titans update across all params (both batches share layout)
__global__ void update_kernel(float* __restrict__ p, float* __restrict__ s,
                              const float* __restrict__ g, long n) {
  const long i = (long)blockIdx.x * blockDim.x + threadIdx.x;
  if (i < n) {
    const float ns = 0.9f * s[i] - 0.01f * g[i];
    s[i] = ns;
    p[i] = 0.99f * p[i] + ns;
  }
}
// z (in gz) *= yt_tail
__global__ void mulz_kernel(float* __restrict__ gz, const float* __restrict__ ytall) {
  const long i = (long)blockIdx.x * blockDim.x + threadIdx.x;
  if (i >= Bt * Lq * Dm) return;
  const long b = i / (Lq * Dm);
  const long rem = i - b * Lq * Dm;
  gz[i] *= ytall[b * SEQ * Dm + TAIL * Dm + rem];
}
// column sums: out[b*outStride + col] = sum_m in[b*inStride + m*N + col]
__global__ void colsum_kernel(float* __restrict__ out, long outStride,
                              const float* __restrict__ in, long inStride,
                              int M, int N) {
  const int col = blockIdx.x * blockDim.x + threadIdx.x;
  const int b = blockIdx.z;
  if (col >= N) return;
  const float* src = in + (size_t)b * inStride;
  float s = 0.0f;
  for (int m = 0; m < M; ++m) s += src[(size_t)m * N + col];
  out[(size_t)b * outStride + col] = s;
}
// per-row LayerNorm (D=512) in place, block=256
__global__ void ln_kernel(float* __restrict__ z, const float* __restrict__ g,
                          const float* __restrict__ beta) {
  __shared__ float red[256];
  float* zr = z + (size_t)blockIdx.x * Dm;
  const int t = threadIdx.x;
  const float x0 = zr[t], x1 = zr[t + 256];
  red[t] = x0 + x1; __syncthreads();
  for (int s2 = 128; s2 > 0; s2 >>= 1) { if (t < s2) red[t] += red[t + s2]; __syncthreads(); }
  const float mu = red[0] * (1.0f / 512.0f);
  __syncthreads();
  const float d0 = x0 - mu, d1 = x1 - mu;
  red[t] = d0 * d0 + d1 * d1; __syncthreads();
  for (int s2 = 128; s2 > 0; s2 >>= 1) { if (t < s2) red[t] += red[t + s2]; __syncthreads(); }
  const float rinv = rsqrtf(red[0] * (1.0f / 512.0f) + 1e-5f);
  zr[t]       = d0 * rinv * g[t]       + beta[t];
  zr[t + 256] = d1 * rinv * g[t + 256] + beta[t + 256];
}

}  // namespace

#define GEMM(TA, TB, ACT, Aptr, lda, sA, Bptr, ldb, sB, biasptr, sBias, Cptr, ldc, sC, M, N, K)   \
  gemm_wmma_f32<TA, TB, ACT><<<dim3((unsigned)((N) / 256), (unsigned)((M) / 16), (unsigned)Bt),   \
                               128, 0, stream>>>(                                                 \
      (Aptr), (int)(lda), (long)(sA), (Bptr), (int)(ldb), (long)(sB), (biasptr), (long)(sBias),   \
      (Cptr), (int)(ldc), (long)(sC), (int)(M), (int)(N), (int)(K))

extern "C" void kernel_launch(void* const* d_in, const int* in_sizes, int n_in,
                              void* d_out, int out_size, void* d_ws, size_t ws_size,
                              hipStream_t stream) {
  const int*   tokens  = (const int*)d_in[0];
  const float* tok_emb = (const float*)d_in[1];
  const float* pos_emb = (const float*)d_in[2];
  const float* Pm      = (const float*)d_in[3];
  const float* Wq_m    = (const float*)d_in[4];
  const float* Wk_m    = (const float*)d_in[5];
  const float* Wv_m    = (const float*)d_in[6];
  const float* W1      = (const float*)d_in[7];
  const float* b1      = (const float*)d_in[8];
  const float* W2      = (const float*)d_in[9];
  const float* b2      = (const float*)d_in[10];
  const float* Wqkv    = (const float*)d_in[11];
  const float* Wo      = (const float*)d_in[12];
  const float* Wg      = (const float*)d_in[13];
  const float* bg      = (const float*)d_in[14];
  const float* ln_g    = (const float*)d_in[15];
  const float* ln_b    = (const float*)d_in[16];
  float* out = (float*)d_out;

  float* ws    = (float*)d_ws;
  float* X     = ws + OFF_X;
  float* par   = ws + OFF_PAR;
  float* sur   = ws + OFF_SUR;
  float* grd   = ws + OFF_GRD;
  float* qm    = ws + OFF_QM;
  float* hr    = ws + OFF_HR;
  float* ht    = ws + OFF_HT;
  float* tt    = ws + OFF_TT;
  float* qkv   = ws + OFF_QKV;
  float* ao    = ws + OFF_AO;
  float* km    = ws + OFF_KM;
  float* vm    = ws + OFF_VM;
  float* hbuf  = ws + OFF_H;
  float* abuf  = ws + OFF_A;
  float* rbuf  = ws + OFF_R;
  float* gabuf = ws + OFF_GA;
  float* qy    = ws + OFF_QY;
  float* hy    = ws + OFF_HY;
  float* ytall = ws + OFF_YT;
  float* mtall = ws + OFF_MT;
  float* gz    = ws + OFF_GZ;

  const float* nob = (const float*)nullptr;

  // ---- init: per-batch memory params <- (W1,b1,W2,b2); surprise <- 0 ----
  for (int b = 0; b < (int)Bt; ++b) {
    float* pb = par + (long)b * PSTR;
    hipMemcpyAsync(pb + PO_W1, W1, Dm * MHh * sizeof(float), hipMemcpyDeviceToDevice, stream);
    hipMemcpyAsync(pb + PO_B1, b1, MHh * sizeof(float),      hipMemcpyDeviceToDevice, stream);
    hipMemcpyAsync(pb + PO_W2, W2, MHh * Dm * sizeof(float), hipMemcpyDeviceToDevice, stream);
    hipMemcpyAsync(pb + PO_B2, b2, Dm * sizeof(float),       hipMemcpyDeviceToDevice, stream);
  }
  zero_kernel<<<(unsigned)((Bt * PSTR + 255) / 256), 256, 0, stream>>>(sur, Bt * PSTR);

  // ---- embedding ----
  embed_kernel<<<dim3((unsigned)Lq, (unsigned)Bt), 256, 0, stream>>>(tokens, tok_emb, pos_emb, X);

  // ---- sequential Titans chunk scan ----
  for (int c = 0; c < (int)NC; ++c) {
    const long co = (long)c * CHc * Dm;   // chunk offset inside X
    const long yo = (long)c * Tt * Dm;    // chunk offset inside ytall/mtall

    // ht = retrieve(chunk_x, params) = silu(chunk_x@Wq_m @ W1p + b1p) @ W2p + b2p
    GEMM(0, 0, 0, X + co, Dm, Lq * Dm, Wq_m, Dm, 0, nob, 0, qm, Dm, CHc * Dm, CHc, Dm, Dm);
    GEMM(0, 0, 1, qm, Dm, CHc * Dm, par + PO_W1, MHh, PSTR, par + PO_B1, PSTR,
         hr, MHh, CHc * MHh, CHc, MHh, Dm);
    GEMM(0, 0, 0, hr, MHh, CHc * MHh, par + PO_W2, Dm, PSTR, par + PO_B2, PSTR,
         ht, Dm, CHc * Dm, CHc, Dm, MHh);

    // t = [P; ht; chunk_x]; yt = attn(t)
    concat_kernel<<<(unsigned)((Bt * Tt * Dm + 255) / 256), 256, 0, stream>>>(tt, Pm, ht, X, c);
    GEMM(0, 0, 0, tt, Dm, Tt * Dm, Wqkv, 3 * Dm, 0, nob, 0, qkv, 3 * Dm, Tt * 3 * Dm, Tt, 3 * Dm, Dm);
    attn_kernel<<<dim3((unsigned)Tt, (unsigned)Hh, (unsigned)Bt), 64, 0, stream>>>(qkv, ao);
    GEMM(0, 0, 0, ao, Dm, Tt * Dm, Wo, Dm, 0, nob, 0, ytall + yo, Dm, SEQ * Dm, Tt, Dm, Dm);

    // inner-loop loss grads on memory MLP
    GEMM(0, 0, 0, ytall + yo, Dm, SEQ * Dm, Wk_m, Dm, 0, nob, 0, km, Dm, Tt * Dm, Tt, Dm, Dm);
    GEMM(0, 0, 0, ytall + yo, Dm, SEQ * Dm, Wv_m, Dm, 0, nob, 0, vm, Dm, Tt * Dm, Tt, Dm, Dm);
    GEMM(0, 0, 0, km, Dm, Tt * Dm, par + PO_W1, MHh, PSTR, par + PO_B1, PSTR,
         hbuf, MHh, Tt * MHh, Tt, MHh, Dm);                            // h (pre-act)
    silu_kernel<<<(unsigned)((Bt * Tt * MHh + 255) / 256), 256, 0, stream>>>(abuf, hbuf, Bt * Tt * MHh);
    GEMM(0, 0, 0, abuf, MHh, Tt * MHh, par + PO_W2, Dm, PSTR, par + PO_B2, PSTR,
         rbuf, Dm, Tt * Dm, Tt, Dm, MHh);                              // o_m
    resid2_kernel<<<(unsigned)((Bt * Tt * Dm + 255) / 256), 256, 0, stream>>>(rbuf, vm, Bt * Tt * Dm);
    GEMM(1, 0, 0, abuf, MHh, Tt * MHh, rbuf, Dm, Tt * Dm, nob, 0,
         grd + PO_W2, Dm, PSTR, MHh, Dm, Tt);                          // gW2 = a^T @ r
    colsum_kernel<<<dim3((unsigned)(Dm / 256), 1, (unsigned)Bt), 256, 0, stream>>>(
        grd + PO_B2, PSTR, rbuf, Tt * Dm, (int)Tt, (int)Dm);           // gb2
    GEMM(0, 1, 0, rbuf, Dm, Tt * Dm, par + PO_W2, Dm, PSTR, nob, 0,
         gabuf, MHh, Tt * MHh, Tt, MHh, Dm);                           // ga = r @ W2p^T
    dsilu_mul_kernel<<<(unsigned)((Bt * Tt * MHh + 255) / 256), 256, 0, stream>>>(gabuf, hbuf, Bt * Tt * MHh);
    GEMM(1, 0, 0, km, Dm, Tt * Dm, gabuf, MHh, Tt * MHh, nob, 0,
         grd + PO_W1, MHh, PSTR, Dm, MHh, Tt);                         // gW1 = km^T @ gh
    colsum_kernel<<<dim3((unsigned)(MHh / 256), 1, (unsigned)Bt), 256, 0, stream>>>(
        grd + PO_B1, PSTR, gabuf, Tt * MHh, (int)Tt, (int)MHh);        // gb1

    // s = 0.9 s - 0.01 g ; p = 0.99 p + s   (all params, both batches)
    update_kernel<<<(unsigned)((Bt * PSTR + 255) / 256), 256, 0, stream>>>(par, sur, grd, Bt * PSTR);

    // Mt_yt = retrieve(yt, new params)
    GEMM(0, 0, 0, ytall + yo, Dm, SEQ * Dm, Wq_m, Dm, 0, nob, 0, qy, Dm, Tt * Dm, Tt, Dm, Dm);
    GEMM(0, 0, 1, qy, Dm, Tt * Dm, par + PO_W1, MHh, PSTR, par + PO_B1, PSTR,
         hy, MHh, Tt * MHh, Tt, MHh, Dm);
    GEMM(0, 0, 0, hy, MHh, Tt * MHh, par + PO_W2, Dm, PSTR, par + PO_B2, PSTR,
         mtall + yo, Dm, SEQ * Dm, Tt, Dm, MHh);
  }

  // ---- tail (last L positions): gate, z, LayerNorm, logits ----
  GEMM(0, 0, 2, mtall + TAIL * Dm, Dm, SEQ * Dm, Wg, Dm, 0, bg, 0,
       gz, Dm, Lq * Dm, Lq, Dm, Dm);                                   // sigmoid(mt@Wg+bg)
  mulz_kernel<<<(unsigned)((Bt * Lq * Dm + 255) / 256), 256, 0, stream>>>(gz, ytall);
  ln_kernel<<<(unsigned)(Bt * Lq), 256, 0, stream>>>(gz, ln_g, ln_b);
  GEMM(0, 1, 0, gz, Dm, Lq * Dm, tok_emb, Dm, 0, nob, 0,
       out, Vv, Lq * Vv, Lq, Vv, Dm);                                  // ot @ tok_emb^T
}